// LocalMultiHeadAttentionModule_33517924778687
// MI455X (gfx1250) — compile-verified
//
#include <hip/hip_runtime.h>
#include <hip/hip_bf16.h>
#include <math.h>

#define NNODES 50000
#define DIN    128
#define DOUT   128          // H*D = 8*16
#define NHEADS 8
#define ROW_TILES (NNODES / 16)       // 3125 (exact)
#define CT_TOTAL  24                   // 3 matrices x 8 col tiles
#define TILES_TOTAL (ROW_TILES * CT_TOTAL)  // 75000 waves

typedef __attribute__((ext_vector_type(2))) float v2f;
typedef __attribute__((ext_vector_type(8))) float v8f;

// ---------------------------------------------------------------------------
// Phase 1: fused Q/K/V projection via V_WMMA_F32_16X16X4_F32.
// One wave computes one 16x16 output tile of one of {Q,K,V}.
// ---------------------------------------------------------------------------
__global__ void __launch_bounds__(256)
qkv_gemm_kernel(const float* __restrict__ X,
                const float* __restrict__ Wq, const float* __restrict__ Wk,
                const float* __restrict__ Wv,
                const float* __restrict__ bq, const float* __restrict__ bk,
                const float* __restrict__ bv,
                float* __restrict__ qo, float* __restrict__ ko,
                float* __restrict__ vo) {
    const int waveId = blockIdx.x * (blockDim.x >> 5) + (threadIdx.x >> 5);
    const int lane   = threadIdx.x & 31;
    if (waveId >= TILES_TOTAL) return;   // whole-wave uniform: EXEC stays all-1s

    const int rowTile  = waveId / CT_TOTAL;
    const int ct       = waveId % CT_TOTAL;
    const int mat      = ct >> 3;            // 0=Q, 1=K, 2=V
    const int colBase  = (ct & 7) << 4;

    const float* W    = (mat == 0) ? Wq : (mat == 1) ? Wk : Wv;
    const float* bias = (mat == 0) ? bq : (mat == 1) ? bk : bv;
    float*       out  = (mat == 0) ? qo : (mat == 1) ? ko : vo;

    const int rowInTile = lane & 15;
    const int kHalf     = (lane >> 4) << 1;   // 0 or 2
    const float* xrow = X + (size_t)(rowTile * 16 + rowInTile) * DIN;
    const float* wcol = W + colBase + (lane & 15);

    v8f acc = {};
#pragma unroll
    for (int kk = 0; kk < DIN; kk += 4) {
        // A: 16x4 f32 tile; lane holds K = kk + kHalf + {0,1} of row (lane&15)
        v2f a;
        a.x = xrow[kk + kHalf + 0];
        a.y = xrow[kk + kHalf + 1];
        // B: 4x16 f32 tile; same K mapping, col = lane&15
        v2f b;
        b.x = wcol[(size_t)(kk + kHalf + 0) * DOUT];
        b.y = wcol[(size_t)(kk + kHalf + 1) * DOUT];
        acc = __builtin_amdgcn_wmma_f32_16x16x4_f32(
            /*neg_a=*/false, a, /*neg_b=*/false, b,
            /*c_mod=*/(short)0, acc, /*reuse_a=*/false, /*reuse_b=*/false);
    }

    const int n = lane & 15;
    const float bcol = bias[colBase + n];
    const int mBase = (lane >> 4) << 3;       // 0 or 8
#pragma unroll
    for (int r = 0; r < 8; ++r) {
        const int m = mBase + r;
        out[(size_t)(rowTile * 16 + m) * DOUT + colBase + n] = acc[r] + bcol;
    }
}

// ---------------------------------------------------------------------------
// Phase 2a: zero accumulators (wV lives in d_out, z in workspace).
// ---------------------------------------------------------------------------
__global__ void zero_kernel(float* __restrict__ out, float* __restrict__ z) {
    const size_t i = (size_t)blockIdx.x * blockDim.x + threadIdx.x;
    if (i < (size_t)NNODES * DOUT) out[i] = 0.0f;
    if (i < (size_t)NNODES * NHEADS) z[i] = 0.0f;
}

// ---------------------------------------------------------------------------
// Phase 2b: per-edge attention. One wave per edge; lane l owns head l/4,
// dims (l%4)*4 .. +3 (a float4). Dot reduced over 4 lanes via shuffles.
// Scatter via f32 atomics (accumulators are L2-resident: 25.6 MB << 192 MB).
// ---------------------------------------------------------------------------
__global__ void __launch_bounds__(256)
edge_kernel(const float* __restrict__ q, const float* __restrict__ k,
            const float* __restrict__ v, const int* __restrict__ src,
            const int* __restrict__ dst, float* __restrict__ wV,
            float* __restrict__ z, int nEdges) {
    const int edge = blockIdx.x * (blockDim.x >> 5) + (threadIdx.x >> 5);
    const int lane = threadIdx.x & 31;
    if (edge >= nEdges) return;           // uniform per wave

    const int s = src[edge];
    const int d = dst[edge];

    const float4 kv = *(const float4*)(k + (size_t)s * DOUT + lane * 4);
    const float4 qv = *(const float4*)(q + (size_t)d * DOUT + lane * 4);
    const float4 vv = *(const float4*)(v + (size_t)s * DOUT + lane * 4);

    float part = kv.x * qv.x + kv.y * qv.y + kv.z * qv.z + kv.w * qv.w;
    // reduce the 4 lanes of this head
    part += __shfl_xor(part, 1, 32);
    part += __shfl_xor(part, 2, 32);

    float sc = part * 0.25f;              // 1/sqrt(D_HEAD=16)
    sc = fminf(fmaxf(sc, -5.0f), 5.0f);
    sc = expf(sc);

    float* wp = wV + (size_t)d * DOUT + lane * 4;
    atomicAdd(wp + 0, vv.x * sc);
    atomicAdd(wp + 1, vv.y * sc);
    atomicAdd(wp + 2, vv.z * sc);
    atomicAdd(wp + 3, vv.w * sc);
    if ((lane & 3) == 0)
        atomicAdd(z + (size_t)d * NHEADS + (lane >> 2), sc);
}

// ---------------------------------------------------------------------------
// Phase 3: out = wV / z  (in place on d_out).
// ---------------------------------------------------------------------------
__global__ void norm_kernel(float* __restrict__ out, const float* __restrict__ z) {
    const size_t i = (size_t)blockIdx.x * blockDim.x + threadIdx.x;
    if (i >= (size_t)NNODES * DOUT) return;
    const size_t node = i / DOUT;
    const size_t head = (i / 16) % NHEADS;
    out[i] = out[i] / z[node * NHEADS + head];
}

extern "C" void kernel_launch(void* const* d_in, const int* in_sizes, int n_in,
                              void* d_out, int out_size, void* d_ws, size_t ws_size,
                              hipStream_t stream) {
    const float* X  = (const float*)d_in[0];
    const float* Wq = (const float*)d_in[1];
    const float* Wk = (const float*)d_in[2];
    const float* Wv = (const float*)d_in[3];
    const float* bq = (const float*)d_in[4];
    const float* bk = (const float*)d_in[5];
    const float* bv = (const float*)d_in[6];
    const int*  src = (const int*)d_in[7];
    const int*  dst = (const int*)d_in[8];
    const int nEdges = in_sizes[7];

    float* out = (float*)d_out;
    float* ws  = (float*)d_ws;
    float* q = ws;
    float* k = ws + (size_t)NNODES * DOUT;
    float* v = ws + (size_t)2 * NNODES * DOUT;
    float* z = ws + (size_t)3 * NNODES * DOUT;   // NNODES*NHEADS floats

    // Phase 1: 75000 waves, 8 waves/block
    const int gemmBlocks = (TILES_TOTAL + 7) / 8;   // 9375
    qkv_gemm_kernel<<<gemmBlocks, 256, 0, stream>>>(X, Wq, Wk, Wv, bq, bk, bv,
                                                    q, k, v);

    // Phase 2a: zero accumulators
    const int nElems = NNODES * DOUT;
    zero_kernel<<<(nElems + 255) / 256, 256, 0, stream>>>(out, z);

    // Phase 2b: one wave per edge
    const int edgeBlocks = (nEdges + 7) / 8;
    edge_kernel<<<edgeBlocks, 256, 0, stream>>>(q, k, v, src, dst, out, z,
                                                nEdges);

    // Phase 3: normalize
    norm_kernel<<<(nElems + 255) / 256, 256, 0, stream>>>(out, z);
}